// OdorClassifier_68839735821021
// MI455X (gfx1250) — compile-verified
//
#include <hip/hip_runtime.h>
#include <hip/hip_bf16.h>
#include <math.h>

typedef __attribute__((ext_vector_type(2))) float v2f;
typedef __attribute__((ext_vector_type(8))) float v8f;

#define B_GR   16384
#define N_NODES 32
#define E_PER   64
#define NT      (B_GR * N_NODES)
#define IN_F    15
#define H1      20
#define H2      155
#define H2P     160
#define MOL     51
#define TASKS   138
#define K1      26
#define K2      21
#define ZW      226
#define ZP      240
#define F1      100
#define F1P     112
#define F2      80
#define OUTP    144
#define BN_EPS  1e-5f

__device__ __forceinline__ float selu_f(float x) {
  const float l = 1.0507009873554805f, a = 1.6732632423543772f;
  return x > 0.f ? l * x : l * a * (expf(x) - 1.f);
}

// ---------------------------------------------------------------- prep kernels
__global__ void k_pad(float* dst, const float* src, int Ns, int Ks, int Np, int Kp) {
  int i = blockIdx.x * blockDim.x + threadIdx.x;
  if (i >= Np * Kp) return;
  int n = i / Kp, k = i - n * Kp;
  dst[i] = (n < Ns && k < Ks) ? src[n * Ks + k] : 0.f;
}

__global__ void k_bnfuse(float* lb, float* sc, float* sh, const float* bias,
                         const float* g, const float* b, const float* m,
                         const float* v, int Ns, int Np) {
  int i = blockIdx.x * blockDim.x + threadIdx.x;
  if (i >= Np) return;
  if (i < Ns) {
    float s = g[i] / sqrtf(v[i] + BN_EPS);
    sc[i] = s; sh[i] = b[i] - m[i] * s; lb[i] = bias ? bias[i] : 0.f;
  } else { sc[i] = 0.f; sh[i] = 0.f; lb[i] = 0.f; }
}

// ---------------------------------------------------------------- stage 1
struct S1Args {
  const float* x; const int* esrc; const int* edst;
  const float* W1; const float* b1;
  const float *bn1g,*bn1b,*bn1m,*bn1v;
  const float* W2; const float* b2;
  const float *bn2g,*bn2b,*bn2m,*bn2v;
  const float *obg,*obb,*obm,*obv;
  const float* Wrel; const float* brel; const float* Wroot;
  float* h2in; float* mask1; float* z;
};

__global__ __launch_bounds__(64) void k_stage1(S1Args A) {
  __shared__ float xs[N_NODES][IN_F];
  __shared__ float agg[N_NODES][H1];
  __shared__ float hh[N_NODES][H1];
  __shared__ float h1s[N_NODES][H1];
  __shared__ float s1[N_NODES];
  __shared__ float mk[N_NODES];
  __shared__ int sl[E_PER], dl[E_PER];
  int g = blockIdx.x, t = threadIdx.x, base = g * N_NODES;

  for (int i = t; i < N_NODES * IN_F; i += 64) xs[i / IN_F][i % IN_F] = A.x[base * IN_F + i];
  if (t < E_PER) { sl[t] = A.esrc[g * E_PER + t] - base; dl[t] = A.edst[g * E_PER + t] - base; }
  for (int i = t; i < N_NODES * H1; i += 64) agg[i / H1][i % H1] = 0.f;
  __syncthreads();
  // agg(x)
  for (int i = t; i < E_PER * IN_F; i += 64) {
    int e = i / IN_F, f = i - e * IN_F;
    atomicAdd(&agg[dl[e]][f], xs[sl[e]][f]);
  }
  __syncthreads();
  // lin1 15->20 + bn + relu
  for (int i = t; i < N_NODES * H1; i += 64) {
    int n = i / H1, o = i - n * H1;
    float acc = A.b1[o];
    for (int k = 0; k < IN_F; ++k) acc += (xs[n][k] + agg[n][k]) * A.W1[o * IN_F + k];
    float s = A.bn1g[o] / sqrtf(A.bn1v[o] + BN_EPS);
    hh[n][o] = fmaxf((acc - A.bn1m[o]) * s + A.bn1b[o], 0.f);
  }
  __syncthreads();
  // lin2 20->20 + bn + relu, then outer bn1 + selu -> h1 ; also re-zero agg
  for (int i = t; i < N_NODES * H1; i += 64) {
    int n = i / H1, o = i - n * H1;
    float acc = A.b2[o];
    for (int k = 0; k < H1; ++k) acc += hh[n][k] * A.W2[o * H1 + k];
    float s = A.bn2g[o] / sqrtf(A.bn2v[o] + BN_EPS);
    acc = fmaxf((acc - A.bn2m[o]) * s + A.bn2b[o], 0.f);
    float so = A.obg[o] / sqrtf(A.obv[o] + BN_EPS);
    h1s[n][o] = selu_f((acc - A.obm[o]) * so + A.obb[o]);
    agg[n][o] = 0.f;
  }
  __syncthreads();
  // agg(h1)
  for (int i = t; i < E_PER * H1; i += 64) {
    int e = i / H1, f = i - e * H1;
    atomicAdd(&agg[dl[e]][f], h1s[sl[e]][f]);
  }
  __syncthreads();
  // pool1 score
  if (t < N_NODES) {
    float acc = A.brel[0];
    for (int k = 0; k < H1; ++k) acc += agg[t][k] * A.Wrel[k] + h1s[t][k] * A.Wroot[k];
    s1[t] = tanhf(acc);
  }
  __syncthreads();
  // stable top-k (rank)
  if (t < N_NODES) {
    float si = s1[t]; int rank = 0;
    for (int j = 0; j < N_NODES; ++j) {
      float sj = s1[j];
      rank += (sj > si) || (sj == si && j < t);
    }
    float m = (rank < K1) ? 1.f : 0.f;
    mk[t] = m;
    A.mask1[base + t] = m;
  }
  __syncthreads();
  // x1p (into hh), re-zero agg
  for (int i = t; i < N_NODES * H1; i += 64) {
    int n = i / H1, o = i - n * H1;
    hh[n][o] = h1s[n][o] * s1[n] * mk[n];
    agg[n][o] = 0.f;
  }
  __syncthreads();
  // r1 -> z[:,0:20]; zero z pad tail; agg(x1p)
  if (t < H1) {
    float acc = 0.f;
    for (int n = 0; n < N_NODES; ++n) acc += hh[n][t];
    A.z[(size_t)g * ZP + t] = acc;
  }
  if (t >= H1 && t < H1 + (ZP - ZW)) A.z[(size_t)g * ZP + ZW + (t - H1)] = 0.f;
  for (int i = t; i < E_PER * H1; i += 64) {
    int e = i / H1, f = i - e * H1;
    atomicAdd(&agg[dl[e]][f], hh[sl[e]][f]);
  }
  __syncthreads();
  // gin2 input = x1p + agg(x1p)
  for (int i = t; i < N_NODES * H1; i += 64) {
    int n = i / H1, o = i - n * H1;
    A.h2in[(size_t)(base + n) * H1 + o] = hh[n][o] + agg[n][o];
  }
}

// ---------------------------------------------------------------- stage 2 (fused gin2, WMMA f32)
struct S2Args {
  const float* h2in;                    // [NT][20]
  const float* W1;                      // [160][20] padded
  const float *lb1,*sc1,*sh1;
  const float* W2;                      // [160][160] padded
  const float *lb2,*s2a,*sh2a,*s2b,*sh2b;
  const float* mask1;
  float* h2;                            // [NT][160]
};

__global__ __launch_bounds__(320) void k_gin2(S2Args A) {
  __shared__ float Y1[16 * H2P];
  int strip = blockIdx.x;
  int lane = threadIdx.x & 31;
  int w = threadIdx.x >> 5;             // 0..9 N-tile
  int rw = lane & 15;
  int kh = (lane >> 4) << 1;            // 0 or 2
  int mh = (lane >> 4) << 3;            // 0 or 8
  int gr = strip * 16 + rw;             // A row for this lane
  int ncol = w * 16 + rw;               // B row / output column
  // ---- lin1: K=20
  const float* Ap = A.h2in + (size_t)gr * H1;
  const float* Bp = A.W1 + (size_t)ncol * H1;
  v8f c = {};
#pragma unroll
  for (int k = 0; k < H1; k += 4) {
    v2f a = *(const v2f*)(Ap + k + kh);
    v2f b = *(const v2f*)(Bp + k + kh);
    c = __builtin_amdgcn_wmma_f32_16x16x4_f32(false, a, false, b, (short)0, c, false, false);
  }
  {
    float lb = A.lb1[ncol], sc = A.sc1[ncol], sh = A.sh1[ncol];
#pragma unroll
    for (int r = 0; r < 8; ++r) {
      float t = c[r] + lb;
      Y1[(r + mh) * H2P + ncol] = fmaxf(t * sc + sh, 0.f);
    }
  }
  __syncthreads();
  // ---- lin2: K=160, A from LDS
  const float* B2 = A.W2 + (size_t)ncol * H2P;
  const float* AL = &Y1[rw * H2P];
  v8f d = {};
#pragma unroll 4
  for (int k = 0; k < H2P; k += 4) {
    v2f a = *(const v2f*)(AL + k + kh);
    v2f b = *(const v2f*)(B2 + k + kh);
    d = __builtin_amdgcn_wmma_f32_16x16x4_f32(false, a, false, b, (short)0, d, false, false);
  }
  {
    float lb = A.lb2[ncol], sa = A.s2a[ncol], sha = A.sh2a[ncol];
    float sb = A.s2b[ncol], shb = A.sh2b[ncol];
#pragma unroll
    for (int r = 0; r < 8; ++r) {
      int grow = strip * 16 + r + mh;
      float t = d[r] + lb;
      t = fmaxf(t * sa + sha, 0.f);        // gin2 bn2 + relu
      t = selu_f(t * sb + shb);            // outer bn2 + selu
      A.h2[(size_t)grow * H2P + ncol] = t * A.mask1[grow];
    }
  }
}

// ---------------------------------------------------------------- stage 3 (pool2 + readout)
struct S3Args {
  const float* h2; const int* esrc; const int* edst; const float* mask1;
  const float* Wrel; const float* brel; const float* Wroot;
  const float* mol; float* z;
};

__global__ __launch_bounds__(128) void k_stage3(S3Args A) {
  __shared__ float hs[N_NODES * H2P];
  __shared__ float ag[N_NODES * H2P];
  __shared__ float s2[N_NODES];
  __shared__ float mk2[N_NODES];
  __shared__ int sl[E_PER], dl[E_PER];
  int g = blockIdx.x, t = threadIdx.x, base = g * N_NODES;
  for (int i = t; i < N_NODES * H2P; i += 128) {
    hs[i] = A.h2[(size_t)base * H2P + i];
    ag[i] = 0.f;
  }
  if (t < E_PER) { sl[t] = A.esrc[g * E_PER + t] - base; dl[t] = A.edst[g * E_PER + t] - base; }
  __syncthreads();
  for (int i = t; i < E_PER * H2; i += 128) {
    int e = i / H2, f = i - e * H2;
    atomicAdd(&ag[dl[e] * H2P + f], hs[sl[e] * H2P + f]);
  }
  __syncthreads();
  if (t < N_NODES) {
    float acc = A.brel[0];
    for (int k = 0; k < H2; ++k) acc += ag[t * H2P + k] * A.Wrel[k] + hs[t * H2P + k] * A.Wroot[k];
    float sc = tanhf(acc);
    mk2[t] = sc;                                           // raw score
    s2[t] = (A.mask1[base + t] > 0.5f) ? sc : -INFINITY;   // candidates = mask1
  }
  __syncthreads();
  if (t < N_NODES) {
    float si = s2[t]; int rank = 0;
    for (int j = 0; j < N_NODES; ++j) {
      float sj = s2[j];
      rank += (sj > si) || (sj == si && j < t);
    }
    mk2[t] = (rank < K2) ? mk2[t] : 0.f;                   // s2 * mask2
  }
  __syncthreads();
  for (int i = t; i < N_NODES * H2P; i += 128) hs[i] *= mk2[i / H2P];
  __syncthreads();
  for (int f = t; f < H2; f += 128) {
    float acc = 0.f;
    for (int n = 0; n < N_NODES; ++n) acc += hs[n * H2P + f];
    A.z[(size_t)g * ZP + H1 + f] = acc;
  }
  for (int f = t; f < MOL; f += 128) A.z[(size_t)g * ZP + H1 + H2 + f] = A.mol[g * MOL + f];
}

// ---------------------------------------------------------------- generic WMMA GEMM (wave = 16x16 tile)
struct GArgs {
  const float* Aa; int lda;
  const float* Bb; int Kp;
  const float *lb,*sc,*sh;
  float* C; int ldc; int Ntrue;
  int Mtiles, Ntiles;
  int act;                                // 0: bias only, 1: bias+bn+relu
};

__global__ __launch_bounds__(256) void k_gemm(GArgs G) {
  int wg = blockIdx.x * (blockDim.x >> 5) + (threadIdx.x >> 5);
  if (wg >= G.Mtiles * G.Ntiles) return;
  int mt = wg / G.Ntiles, nt = wg - mt * G.Ntiles;
  int lane = threadIdx.x & 31;
  int rw = lane & 15;
  int kh = (lane >> 4) << 1;
  int mh = (lane >> 4) << 3;
  int gr = mt * 16 + rw;
  int ncol = nt * 16 + rw;
  const float* Ap = G.Aa + (size_t)gr * G.lda;
  const float* Bp = G.Bb + (size_t)ncol * G.Kp;
  v8f c = {};
  for (int k = 0; k < G.Kp; k += 4) {
    v2f a = *(const v2f*)(Ap + k + kh);
    v2f b = *(const v2f*)(Bp + k + kh);
    c = __builtin_amdgcn_wmma_f32_16x16x4_f32(false, a, false, b, (short)0, c, false, false);
  }
  float lb = G.lb[ncol];
  float scv = 1.f, shv = 0.f;
  if (G.act) { scv = G.sc[ncol]; shv = G.sh[ncol]; }
  bool incol = ncol < G.Ntrue;
#pragma unroll
  for (int r = 0; r < 8; ++r) {
    float t = c[r] + lb;
    if (G.act) t = fmaxf(t * scv + shv, 0.f);
    if (incol) G.C[(size_t)(mt * 16 + r + mh) * G.ldc + ncol] = t;
  }
}

// ---------------------------------------------------------------- host
extern "C" void kernel_launch(void* const* d_in, const int* in_sizes, int n_in,
                              void* d_out, int out_size, void* d_ws, size_t ws_size,
                              hipStream_t stream) {
  const float* x    = (const float*)d_in[0];
  const int*   esrc = (const int*)d_in[1];
  const int    E    = in_sizes[1] / 2;
  const int*   edst = esrc + E;
  const float* mol  = (const float*)d_in[2];
  // params (insertion-order flatten)
  const float* g1l1W = (const float*)d_in[4];
  const float* g1l1b = (const float*)d_in[5];
  const float* g1b1g = (const float*)d_in[6];
  const float* g1b1b = (const float*)d_in[7];
  const float* g1b1m = (const float*)d_in[8];
  const float* g1b1v = (const float*)d_in[9];
  const float* g1l2W = (const float*)d_in[10];
  const float* g1l2b = (const float*)d_in[11];
  const float* g1b2g = (const float*)d_in[12];
  const float* g1b2b = (const float*)d_in[13];
  const float* g1b2m = (const float*)d_in[14];
  const float* g1b2v = (const float*)d_in[15];
  const float* ob1g = (const float*)d_in[16];
  const float* ob1b = (const float*)d_in[17];
  const float* ob1m = (const float*)d_in[18];
  const float* ob1v = (const float*)d_in[19];
  const float* p1Wrel = (const float*)d_in[20];
  const float* p1brel = (const float*)d_in[21];
  const float* p1Wroot = (const float*)d_in[22];
  const float* g2l1W = (const float*)d_in[23];
  const float* g2l1b = (const float*)d_in[24];
  const float* g2b1g = (const float*)d_in[25];
  const float* g2b1b = (const float*)d_in[26];
  const float* g2b1m = (const float*)d_in[27];
  const float* g2b1v = (const float*)d_in[28];
  const float* g2l2W = (const float*)d_in[29];
  const float* g2l2b = (const float*)d_in[30];
  const float* g2b2g = (const float*)d_in[31];
  const float* g2b2b = (const float*)d_in[32];
  const float* g2b2m = (const float*)d_in[33];
  const float* g2b2v = (const float*)d_in[34];
  const float* ob2g = (const float*)d_in[35];
  const float* ob2b = (const float*)d_in[36];
  const float* ob2m = (const float*)d_in[37];
  const float* ob2v = (const float*)d_in[38];
  const float* p2Wrel = (const float*)d_in[39];
  const float* p2brel = (const float*)d_in[40];
  const float* p2Wroot = (const float*)d_in[41];
  const float* fc1W = (const float*)d_in[42];
  const float* fc1b = (const float*)d_in[43];
  const float* mb1g = (const float*)d_in[44];
  const float* mb1b = (const float*)d_in[45];
  const float* mb1m = (const float*)d_in[46];
  const float* mb1v = (const float*)d_in[47];
  const float* fc2W = (const float*)d_in[48];
  const float* fc2b = (const float*)d_in[49];
  const float* mb2g = (const float*)d_in[50];
  const float* mb2b = (const float*)d_in[51];
  const float* mb2m = (const float*)d_in[52];
  const float* mb2v = (const float*)d_in[53];
  const float* outW = (const float*)d_in[54];
  const float* outb = (const float*)d_in[55];

  // workspace layout (float, 64-element aligned)
  float* w = (float*)d_ws;
  size_t off = 0;
  auto alloc = [&](size_t n) { float* p = w + off; off += (n + 63) & ~(size_t)63; return p; };
  float* h2in  = alloc((size_t)NT * H1);
  float* h2    = alloc((size_t)NT * H2P);
  float* mask1 = alloc(NT);
  float* z     = alloc((size_t)B_GR * ZP);
  float* a1    = alloc((size_t)B_GR * F1P);
  float* a2    = alloc((size_t)B_GR * F2);
  float* Wg2a  = alloc(H2P * H1);
  float* Wg2b  = alloc(H2P * H2P);
  float* Wfc1  = alloc(F1P * ZP);
  float* Wfc2  = alloc(F2 * F1P);
  float* Wout  = alloc(OUTP * F2);
  float* lb1 = alloc(H2P); float* sc1 = alloc(H2P); float* sh1 = alloc(H2P);
  float* lb2 = alloc(H2P); float* s2a = alloc(H2P); float* sh2a = alloc(H2P);
  float* lbx = alloc(H2P); float* s2b = alloc(H2P); float* sh2b = alloc(H2P);
  float* lbf1 = alloc(F1P); float* scf1 = alloc(F1P); float* shf1 = alloc(F1P);
  float* lbf2 = alloc(F2);  float* scf2 = alloc(F2);  float* shf2 = alloc(F2);
  float* lbo = alloc(OUTP);
  (void)n_in; (void)out_size; (void)ws_size;

  // ---- prep: padded weights + fused bn constants
  auto gpad = [&](float* dst, const float* src, int Ns, int Ks, int Np, int Kp) {
    int tot = Np * Kp;
    hipLaunchKernelGGL(k_pad, dim3((tot + 255) / 256), dim3(256), 0, stream, dst, src, Ns, Ks, Np, Kp);
  };
  gpad(Wg2a, g2l1W, H2, H1, H2P, H1);
  gpad(Wg2b, g2l2W, H2, H2, H2P, H2P);
  gpad(Wfc1, fc1W, F1, ZW, F1P, ZP);
  gpad(Wfc2, fc2W, F2, F1, F2, F1P);
  gpad(Wout, outW, TASKS, F2, OUTP, F2);
  gpad(lbo, outb, TASKS, 1, OUTP, 1);
  auto gbn = [&](float* lb, float* sc, float* sh, const float* bias,
                 const float* g, const float* b, const float* m, const float* v,
                 int Ns, int Np) {
    hipLaunchKernelGGL(k_bnfuse, dim3((Np + 255) / 256), dim3(256), 0, stream,
                       lb, sc, sh, bias, g, b, m, v, Ns, Np);
  };
  gbn(lb1, sc1, sh1, g2l1b, g2b1g, g2b1b, g2b1m, g2b1v, H2, H2P);
  gbn(lb2, s2a, sh2a, g2l2b, g2b2g, g2b2b, g2b2m, g2b2v, H2, H2P);
  gbn(lbx, s2b, sh2b, nullptr, ob2g, ob2b, ob2m, ob2v, H2, H2P);
  gbn(lbf1, scf1, shf1, fc1b, mb1g, mb1b, mb1m, mb1v, F1, F1P);
  gbn(lbf2, scf2, shf2, fc2b, mb2g, mb2b, mb2m, mb2v, F2, F2);

  // ---- stage 1
  S1Args s1{x, esrc, edst, g1l1W, g1l1b, g1b1g, g1b1b, g1b1m, g1b1v,
            g1l2W, g1l2b, g1b2g, g1b2b, g1b2m, g1b2v,
            ob1g, ob1b, ob1m, ob1v, p1Wrel, p1brel, p1Wroot,
            h2in, mask1, z};
  hipLaunchKernelGGL(k_stage1, dim3(B_GR), dim3(64), 0, stream, s1);

  // ---- stage 2: fused gin2 with f32 WMMA
  S2Args s2{h2in, Wg2a, lb1, sc1, sh1, Wg2b, lb2, s2a, sh2a, s2b, sh2b, mask1, h2};
  hipLaunchKernelGGL(k_gin2, dim3(NT / 16), dim3(320), 0, stream, s2);

  // ---- stage 3: pool2 + readout
  S3Args s3{h2, esrc, edst, mask1, p2Wrel, p2brel, p2Wroot, mol, z};
  hipLaunchKernelGGL(k_stage3, dim3(B_GR), dim3(128), 0, stream, s3);

  // ---- MLP head (WMMA GEMMs)
  auto gemm = [&](const float* Aa, int lda, const float* Bb, int Kp,
                  const float* lb, const float* sc, const float* sh,
                  float* C, int ldc, int Ntrue, int Ntiles, int act) {
    GArgs ga{Aa, lda, Bb, Kp, lb, sc, sh, C, ldc, Ntrue, B_GR / 16, Ntiles, act};
    int total = (B_GR / 16) * Ntiles;
    hipLaunchKernelGGL(k_gemm, dim3((total + 7) / 8), dim3(256), 0, stream, ga);
  };
  gemm(z, ZP, Wfc1, ZP, lbf1, scf1, shf1, a1, F1P, F1P, F1P / 16, 1);
  gemm(a1, F1P, Wfc2, F1P, lbf2, scf2, shf2, a2, F2, F2, F2 / 16, 1);
  gemm(a2, F2, Wout, F2, lbo, lbo, lbo, (float*)d_out, TASKS, TASKS, OUTP / 16, 0);
}